// GraphConvolution_43602507989463
// MI455X (gfx1250) — compile-verified
//
#include <hip/hip_runtime.h>
#include <hip/hip_bf16.h>

#define DD 128

typedef __attribute__((ext_vector_type(2))) float v2f;
typedef __attribute__((ext_vector_type(8))) float v8f;

// ---------------------------------------------------------------------------
// Kernel 1: zero the aggregation buffer (d_out doubles as aggr staging).
// ---------------------------------------------------------------------------
__global__ void __launch_bounds__(256)
gc_zero_kernel(float4* __restrict__ p, int n4) {
    int i = blockIdx.x * blockDim.x + threadIdx.x;
    if (i < n4) p[i] = make_float4(0.f, 0.f, 0.f, 0.f);
}

// ---------------------------------------------------------------------------
// Kernel 2: edge scatter-add. One wave32 per edge: lane L loads float4
// x[src][4L..4L+3] (coalesced 512B/wave) and atomically adds into
// aggr[dst][4L..4L+3]. aggr (25.6 MB) is L2-resident on MI455X (192 MB L2),
// so the 102M fp32 atomics stay on-chip.
// ---------------------------------------------------------------------------
__global__ void __launch_bounds__(256)
gc_scatter_kernel(const float* __restrict__ x,
                  const int* __restrict__ esrc,
                  const int* __restrict__ edst,
                  float* aggr, int n_edges) {
    int wave = (blockIdx.x * blockDim.x + threadIdx.x) >> 5;
    int lane = threadIdx.x & 31;
    if (wave >= n_edges) return;
    int s = esrc[wave];   // wave-uniform -> scalarized
    int d = edst[wave];
    const float4* srcv = (const float4*)(x + (size_t)s * DD);
    float4 v = srcv[lane];
    float* dp = aggr + (size_t)d * DD + lane * 4;
    atomicAdd(dp + 0, v.x);
    atomicAdd(dp + 1, v.y);
    atomicAdd(dp + 2, v.z);
    atomicAdd(dp + 3, v.w);
}

// ---------------------------------------------------------------------------
// Kernel 3: fused  out = x@W1^T + aggr@W2^T + (b1+b2)  via
// V_WMMA_F32_16X16X4_F32.  Block = 256 threads = 8 waves; wave w owns node
// rows [blk*128 + 16w, +16) x all 128 output features (8 16x16 D tiles,
// v8f each).  A-fragment layout per ISA (32-bit A 16x4): lanes 0-15 hold
// {K=k,k+1} of row (lane), lanes 16-31 hold {K=k+2,k+3} of row (lane-16).
// B[k][n] = W[n][k] (torch Linear), so the B-fragment is the same contiguous
// float2 read from row n of W.
//
// Key scheduling fix vs previous round: all 18 fragment loads of a k-step go
// into DISTINCT temporaries before the 16 chained WMMAs, so the backend
// issues the loads as a batch (one s_wait_loadcnt per 16 WMMAs instead of
// per WMMA) and, with unroll-2, overlaps the next step's loads with the
// current step's matrix ops.
//
// aggr is read from d_out and out written back in place; every store depends
// on every load through the WMMA chain, and rows are wave-exclusive, so the
// in-place update is hazard-free.
// ---------------------------------------------------------------------------
__global__ void __launch_bounds__(256)
gc_gemm_kernel(const float* __restrict__ x,
               const float* __restrict__ W1,
               const float* __restrict__ b1,
               const float* __restrict__ W2,
               const float* __restrict__ b2,
               const float* aggr,   // == out (in place)
               float* out,
               int n_nodes) {
    const int lane = threadIdx.x & 31;
    const int wave = threadIdx.x >> 5;
    const int r16  = lane & 15;
    const int half = lane >> 4;            // 0: K pair {0,1}; 1: K pair {2,3}
    const int row_base = blockIdx.x * 128 + wave * 16;

    // Clamp load row so EXEC stays all-1s for WMMA (stores are predicated).
    int arow = row_base + r16;
    int arow_c = (arow < n_nodes) ? arow : (n_nodes - 1);
    const float* xrow = x    + (size_t)arow_c * DD;
    const float* grow = aggr + (size_t)arow_c * DD;

    // Per-lane weight base: row r16 of W; nt advances by immediate offset
    // nt*16*DD*4 = nt*8192 bytes (folds into the load's inst offset).
    const float* w1p = W1 + (size_t)r16 * DD;
    const float* w2p = W2 + (size_t)r16 * DD;

    // Bias-initialized accumulators: acc[nt] lane (0-15) -> N = nt*16+lane,
    // constant across the 8 VGPRs, so broadcast b1[n]+b2[n].
    v8f acc[8];
#pragma unroll
    for (int nt = 0; nt < 8; ++nt) {
        float bv = b1[nt * 16 + r16] + b2[nt * 16 + r16];
        acc[nt] = (v8f){bv, bv, bv, bv, bv, bv, bv, bv};
    }

#pragma unroll 2
    for (int k = 0; k < DD; k += 4) {
        const int kc = k + half * 2;                 // 8B-aligned

        // ---- batched fragment fetch (distinct registers) ----
        v2f ax = *(const v2f*)(xrow + kc);           // A frag (x)
        v2f ag = *(const v2f*)(grow + kc);           // A frag (aggr)
        v2f bw1[8], bw2[8];
#pragma unroll
        for (int nt = 0; nt < 8; ++nt) {
            bw1[nt] = *(const v2f*)(w1p + nt * 16 * DD + kc);
            bw2[nt] = *(const v2f*)(w2p + nt * 16 * DD + kc);
        }

        // ---- 16 chained WMMAs, no loads in between ----
#pragma unroll
        for (int nt = 0; nt < 8; ++nt) {
            acc[nt] = __builtin_amdgcn_wmma_f32_16x16x4_f32(
                false, ax, false, bw1[nt], (short)0, acc[nt], false, false);
            acc[nt] = __builtin_amdgcn_wmma_f32_16x16x4_f32(
                false, ag, false, bw2[nt], (short)0, acc[nt], false, false);
        }
    }

    // D layout: VGPR r, lanes 0-15 -> (M=r, N=lane); lanes 16-31 -> (M=r+8).
#pragma unroll
    for (int r = 0; r < 8; ++r) {
        int m = row_base + r + half * 8;
        if (m < n_nodes) {
            float* orow = out + (size_t)m * DD + r16;
#pragma unroll
            for (int nt = 0; nt < 8; ++nt) {
                orow[nt * 16] = acc[nt][r];
            }
        }
    }
}

// ---------------------------------------------------------------------------
extern "C" void kernel_launch(void* const* d_in, const int* in_sizes, int n_in,
                              void* d_out, int out_size, void* d_ws, size_t ws_size,
                              hipStream_t stream) {
    const float* x    = (const float*)d_in[0];
    const float* W1   = (const float*)d_in[1];
    const float* b1   = (const float*)d_in[2];
    const float* W2   = (const float*)d_in[3];
    const float* b2   = (const float*)d_in[4];
    const int*   esrc = (const int*)d_in[5];
    const int*   edst = (const int*)d_in[6];
    float*       out  = (float*)d_out;

    const int n_nodes = in_sizes[0] / DD;     // 50000
    const int n_edges = in_sizes[5];          // 800000

    // 1) zero aggr staging area (d_out)
    {
        int n4 = (n_nodes * DD) / 4;
        int blocks = (n4 + 255) / 256;
        gc_zero_kernel<<<blocks, 256, 0, stream>>>((float4*)out, n4);
    }
    // 2) edge scatter-add into d_out
    {
        int waves_per_block = 256 / 32;
        int blocks = (n_edges + waves_per_block - 1) / waves_per_block;
        gc_scatter_kernel<<<blocks, 256, 0, stream>>>(x, esrc, edst, out, n_edges);
    }
    // 3) fused dual fp32 WMMA GEMM + bias, in place over d_out
    {
        int blocks = (n_nodes + 127) / 128;
        gc_gemm_kernel<<<blocks, 256, 0, stream>>>(x, W1, b1, W2, b2, out, out, n_nodes);
    }
}